// FineMatching_76381698392657
// MI455X (gfx1250) — compile-verified
//
#include <hip/hip_runtime.h>
#include <cstdint>

// FineMatching for MI455X (gfx1250).
// P=512 proposals, each a 128x128 f32 tile. Memory-bound (~96MB traffic,
// ~4.1us floor @ 23.3 TB/s). Tiles are streamed into LDS with the Tensor
// Data Mover (double-buffered, TENSORcnt-synced), with TDM's LDS-padding
// feature inserting 1 dword per 128 dwords so the LDS tile has a 129-dword
// row stride => bank-conflict-free row scans AND column scans.
// Top-3 selection is done on raw scores (exp is monotonic); exp() is only
// applied at output time.

#define P_N 512
#define R_N 128
#define S_N 128
#define THRESH 0.05f

#define NUM_WG 128
#define PER_WG (P_N / NUM_WG)        // 4 proposals per workgroup
#define ROW_STRIDE 129               // padded LDS row stride in dwords (TDM pad)
#define TILE_DW (R_N * ROW_STRIDE + 4)

typedef unsigned int u32x4 __attribute__((ext_vector_type(4)));
typedef int          i32x8 __attribute__((ext_vector_type(8)));
typedef int          i32x4 __attribute__((ext_vector_type(4)));

// Issue a TDM load of one 128x128 f32 tile (row-major, stride 128) from
// global memory into LDS at lds_byte_off, padded to a 129-dword row stride.
// D# layout per CDNA5 ISA ch.8 (group0: count/lds/global/type, group1:
// dims/strides/pad). Groups 2/3 zero (2-D tensor). Wave-level op: caller
// must ensure only one wave executes this.
__device__ __forceinline__ void tdm_load_tile(const float* gsrc, unsigned lds_byte_off) {
  unsigned long long ga = (unsigned long long)(uintptr_t)gsrc;
  u32x4 g0;
  g0[0] = 1u;                                   // count=1 valid descriptor, user mode
  g0[1] = lds_byte_off;                         // lds_addr
  g0[2] = (unsigned)(ga & 0xFFFFFFFFull);       // global_addr[31:0]
  g0[3] = (unsigned)((ga >> 32) & 0x01FFFFFFull)// global_addr[56:32]
        | (2u << 30);                           // type=2 ("image")
  i32x8 g1;
  g1[0] = (int)((2u << 16)                      // data_size: 4 bytes
              | (1u << 20)                      // pad_enable
              | (6u << 22)                      // pad_interval: 128 dwords
              | (0u << 25));                    // pad_amount: 1 dword
  g1[1] = (int)(128u << 16);                    // tensor_dim0 = 128 (bits 79:48, lo16)
  g1[2] = (int)(128u << 16);                    // dim0 hi=0 | tensor_dim1 = 128 (lo16)
  g1[3] = (int)(128u << 16);                    // dim1 hi=0 | tile_dim0 = 128
  g1[4] = 128;                                  // tile_dim1 = 128, tile_dim2 = 0
  g1[5] = 128;                                  // tensor_dim0_stride = 128 elements
  g1[6] = 0;
  g1[7] = 0;
  i32x4 gz4 = {0, 0, 0, 0};
  i32x8 gz8 = {0, 0, 0, 0, 0, 0, 0, 0};
  // clang-23 / therock-10.0 lane: 6-arg form
  // (u32x4 g0, i32x8 g1, i32x4 g2, i32x4 g3, i32x8 extra, i32 cpol)
  __builtin_amdgcn_tensor_load_to_lds(g0, g1, gz4, gz4, gz8, 0);
}

__global__ __launch_bounds__(256, 2)
void fine_matching_kernel(const float* __restrict__ msm,
                          const int*   __restrict__ refMask,
                          const int*   __restrict__ srcMask,
                          const float* __restrict__ nodeScore,
                          float*       __restrict__ out) {
  __shared__ float tile[2][TILE_DW];   // double-buffered padded tiles
  __shared__ int   rtIdx[R_N * 3];     // row top-3 column indices
  __shared__ int   ctIdx[S_N * 3];     // col top-3 row indices
  __shared__ int   refM[R_N];
  __shared__ int   srcM[S_N];

  const int tid   = threadIdx.x;
  const int wave  = tid >> 5;
  const int pbase = blockIdx.x * PER_WG;

  float* outScore = out;
  float* outCorr  = out + (size_t)P_N * R_N * S_N;

  // Prime the pipeline: DMA tile 0 into buffer 0.
  if (wave == 0) {
    tdm_load_tile(msm + (size_t)pbase * (R_N * S_N),
                  (unsigned)(uintptr_t)&tile[0][0]);
  }

  for (int j = 0; j < PER_WG; ++j) {
    const int p = pbase + j;
    float* cur = tile[j & 1];

    if (wave == 0) {
      if (j + 1 < PER_WG) {
        // Prefetch next tile into the other buffer, then wait for current.
        tdm_load_tile(msm + (size_t)(p + 1) * (R_N * S_N),
                      (unsigned)(uintptr_t)&tile[(j + 1) & 1][0]);
        __builtin_amdgcn_s_wait_tensorcnt(1);   // oldest (tile j) complete
      } else {
        __builtin_amdgcn_s_wait_tensorcnt(0);
      }
    }
    // Stage masks while wave0 handles the DMA handshake.
    if (tid < R_N) refM[tid]        = refMask[(size_t)p * R_N + tid];
    else           srcM[tid - R_N]  = srcMask[(size_t)p * S_N + (tid - R_N)];
    __syncthreads();   // tile j visible to all waves; masks staged

    // ---- Phase A: per-row and per-column top-3 on RAW scores ----
    // Lanes map to distinct rows/cols; 129-dword stride => conflict-free.
    if (tid < R_N) {
      const int r = tid;
      const float* row = &cur[r * ROW_STRIDE];
      float b0 = -INFINITY, b1 = -INFINITY, b2 = -INFINITY;
      int   j0 = 0, j1 = 0, j2 = 0;
      for (int s = 0; s < S_N; ++s) {
        float v = row[s];
        if (v > b0)      { b2 = b1; j2 = j1; b1 = b0; j1 = j0; b0 = v; j0 = s; }
        else if (v > b1) { b2 = b1; j2 = j1; b1 = v;  j1 = s; }
        else if (v > b2) { b2 = v;  j2 = s; }
      }
      rtIdx[r * 3 + 0] = j0; rtIdx[r * 3 + 1] = j1; rtIdx[r * 3 + 2] = j2;
    } else {
      const int c = tid - R_N;
      float b0 = -INFINITY, b1 = -INFINITY, b2 = -INFINITY;
      int   j0 = 0, j1 = 0, j2 = 0;
      for (int r = 0; r < R_N; ++r) {
        float v = cur[r * ROW_STRIDE + c];
        if (v > b0)      { b2 = b1; j2 = j1; b1 = b0; j1 = j0; b0 = v; j0 = r; }
        else if (v > b1) { b2 = b1; j2 = j1; b1 = v;  j1 = r; }
        else if (v > b2) { b2 = v;  j2 = r; }
      }
      ctIdx[c * 3 + 0] = j0; ctIdx[c * 3 + 1] = j1; ctIdx[c * 3 + 2] = j2;
    }
    __syncthreads();

    // ---- Phase B: emit score_map and corr_map, coalesced float4 stores ----
    const float ns = nodeScore[p];
    float4* os = (float4*)(outScore + (size_t)p * (R_N * S_N));
    float4* oc = (float4*)(outCorr  + (size_t)p * (R_N * S_N));

    #pragma unroll 4
    for (int i = 0; i < 16; ++i) {
      const int q    = i * 256 + tid;     // float4 index within tile
      const int base = q << 2;            // element index
      const int r    = base >> 7;
      const int s0   = base & (S_N - 1);
      const int a0 = rtIdx[r * 3 + 0];
      const int a1 = rtIdx[r * 3 + 1];
      const int a2 = rtIdx[r * 3 + 2];
      const bool rm = (refM[r] != 0);
      float scv[4], cov[4];
      #pragma unroll
      for (int k = 0; k < 4; ++k) {
        const int s   = s0 + k;
        const float raw = cur[r * ROW_STRIDE + s];
        const bool inRow = (s == a0) | (s == a1) | (s == a2);
        const bool inCol = (r == ctIdx[s * 3 + 0]) |
                           (r == ctIdx[s * 3 + 1]) |
                           (r == ctIdx[s * 3 + 2]);
        const float ev = __expf(raw);     // only meaningful when member
        const float rv = inRow ? ev : 0.0f;
        const float sv = inCol ? ev : 0.0f;
        const bool  mk = rm & (srcM[s] != 0);
        scv[k] = (rv + sv) * 0.5f * ns;
        cov[k] = (mk & ((rv > THRESH) | (sv > THRESH))) ? 1.0f : 0.0f;
      }
      float4 sc, co;
      sc.x = scv[0]; sc.y = scv[1]; sc.z = scv[2]; sc.w = scv[3];
      co.x = cov[0]; co.y = cov[1]; co.z = cov[2]; co.w = cov[3];
      os[q] = sc;
      oc[q] = co;
    }
    __syncthreads();   // all waves done with 'cur' before it is re-filled
  }
}

extern "C" void kernel_launch(void* const* d_in, const int* in_sizes, int n_in,
                              void* d_out, int out_size, void* d_ws, size_t ws_size,
                              hipStream_t stream) {
  const float* msm  = (const float*)d_in[0];   // [512,128,128] f32
  const int*   refm = (const int*)d_in[1];     // [512,128] bool->int
  const int*   srcm = (const int*)d_in[2];     // [512,128] bool->int
  const float* nsc  = (const float*)d_in[3];   // [512] f32
  float* outp = (float*)d_out;                 // score_map | corr_map (flat)

  fine_matching_kernel<<<dim3(NUM_WG), dim3(256), 0, stream>>>(
      msm, refm, srcm, nsc, outp);
}